// TemperatureModel_81767587381683
// MI455X (gfx1250) — compile-verified
//
#include <hip/hip_runtime.h>

// TemperatureModel: out[n, m] = (m == argmin_k |means[k]-target[n]|) ? means[m] : 0
// N = 1,048,576 targets, M = 256 means. Output = 1 GiB f32 -> store-bandwidth bound
// (~46 us @ 23.3 TB/s). Score matrix computed on the matrix pipe via
// V_WMMA_F32_16X16X4_F32: (m-t)^2 + 1 = m^2*1 + (-2m)*t + 1*t^2 + C(=1);
// the compiler encodes C as inline constant 1.0, so the positivity bias is free.

typedef __attribute__((ext_vector_type(2))) float v2f;
typedef __attribute__((ext_vector_type(4))) float v4f;
typedef __attribute__((ext_vector_type(8))) float v8f;

#define M_MEANS        256
#define ROWS_PER_BLOCK 256
#define THREADS        256

__global__ __launch_bounds__(THREADS)
void temp_argmin_scatter(const float* __restrict__ target,
                         const float* __restrict__ means,
                         float* __restrict__ out, int N)
{
    __shared__ float lds_means[M_MEANS];
    __shared__ float lds_t[ROWS_PER_BLOCK];
    __shared__ int   lds_bidx[ROWS_PER_BLOCK];
    __shared__ float lds_bval[ROWS_PER_BLOCK];

    const int tid     = threadIdx.x;
    const int rowBase = blockIdx.x * ROWS_PER_BLOCK;
    if (rowBase >= N) return;

    // Stage 0: means (reused by every block -> regular temporal, stays in L2)
    // and this block's 256 targets (read exactly once globally -> non-temporal).
    lds_means[tid] = means[tid];
    lds_t[tid]     = __builtin_nontemporal_load(&target[rowBase + tid]);
    __syncthreads();

    const int lane = tid & 31;
    const int wave = tid >> 5;
    const int half = lane >> 4;   // 0: lanes 0-15, 1: lanes 16-31
    const int l16  = lane & 15;

    const v8f cones = {1.0f, 1.0f, 1.0f, 1.0f, 1.0f, 1.0f, 1.0f, 1.0f};

    // Stage 1: each wave handles two groups of 16 targets. For each group,
    // 16 WMMA tiles sweep all 256 means. D-matrix layout (16x16 f32 C/D):
    // lane L holds column N = L%16 (its target), rows M = vgpr + 8*(L>=16).
    for (int p = 0; p < 2; ++p) {
        const int group = (wave << 1) | p;                 // 0..15
        const float t = lds_t[group * 16 + l16];

        // B (4x16, K rows x N cols), 2 VGPRs: vgpr holds K = vgpr + 2*(lane>=16).
        // B[:,n] = [1, t, t^2, 0]
        v2f B;
        B.x = half ? t * t : 1.0f;   // K=0 row (ones) / K=2 row (t^2)
        B.y = half ? 0.0f  : t;      // K=1 row (t)    / K=3 row (zeros)

        unsigned best = 0xFFFFFFFFu;
        #pragma unroll
        for (int tile = 0; tile < 16; ++tile) {
            const float m = lds_means[tile * 16 + l16];
            // A (16x4), 2 VGPRs: lane L row M = L%16, K = vgpr + 2*(lane>=16).
            // A[m,:] = [m^2, -2m, 1, 0]
            v2f A;
            A.x = half ? 1.0f : m * m;
            A.y = half ? 0.0f : -2.0f * m;

            v8f d = __builtin_amdgcn_wmma_f32_16x16x4_f32(
                false, A, false, B, (short)0, cones, false, false);

            // Pack (score | mean_index) into one u32 key. Score is positive
            // (bias via C), so IEEE order == u32 order. The packed low byte is
            // a COMPILE-TIME constant (tile*16 + v, bit 3 always 0); the lane's
            // half-bit (+8) is OR'd into `best` once after the sweep, which is
            // order-preserving since it is identical across all of this lane's
            // keys. Lower index still wins ties, matching jnp.argmin.
            #pragma unroll
            for (int v = 0; v < 8; ++v) {
                unsigned key = (__float_as_uint(d[v]) & 0xFFFFFF00u)
                             | (unsigned)(tile * 16 + v);
                best = best < key ? best : key;
            }
        }
        // Embed this lane's M-range offset (rows M = v + 8*half) into the index
        // byte, then merge the two half-wave M-ranges: SWAPX16 via ds_swizzle
        // (xor_mask=0x10, and_mask=0x1F).
        best |= (unsigned)(half << 3);
        unsigned other = (unsigned)__builtin_amdgcn_ds_swizzle((int)best, 0x401f);
        best = best < other ? best : other;

        if (half == 0) {
            int n  = group * 16 + l16;
            int bi = (int)(best & 0xFFu);
            lds_bidx[n] = bi;
            lds_bval[n] = lds_means[bi];
        }
    }
    __syncthreads();

    // Stage 2: emit 256 rows x 1 KiB of output. 256 threads x b128 stores,
    // fully coalesced (a wave covers 512 contiguous bytes of one row).
    // Non-temporal: 1 GiB output >> 192 MB L2, don't thrash it.
    float* blockOut = out + (size_t)rowBase * M_MEANS;
    #pragma unroll 4
    for (int i = tid; i < ROWS_PER_BLOCK * (M_MEANS / 4); i += THREADS) {
        int row   = i >> 6;        // 0..255
        int chunk = i & 63;        // 16B chunk within the row
        int bi    = lds_bidx[row];
        float bv  = lds_bval[row];
        int cbase = chunk << 2;
        v4f o;
        o.x = (bi == cbase + 0) ? bv : 0.0f;
        o.y = (bi == cbase + 1) ? bv : 0.0f;
        o.z = (bi == cbase + 2) ? bv : 0.0f;
        o.w = (bi == cbase + 3) ? bv : 0.0f;
        __builtin_nontemporal_store(o, (v4f*)(blockOut + (size_t)row * M_MEANS + cbase));
    }
}

extern "C" void kernel_launch(void* const* d_in, const int* in_sizes, int n_in,
                              void* d_out, int out_size, void* d_ws, size_t ws_size,
                              hipStream_t stream) {
    const float* target = (const float*)d_in[0];
    const float* means  = (const float*)d_in[1];
    float* out          = (float*)d_out;
    const int N = in_sizes[0];                 // 1,048,576
    const int blocks = N / ROWS_PER_BLOCK;     // 4096
    temp_argmin_scatter<<<blocks, THREADS, 0, stream>>>(target, means, out, N);
}